// EquivariantDecompositionModule_42279658062201
// MI455X (gfx1250) — compile-verified
//
#include <hip/hip_runtime.h>
#include <hip/hip_bf16.h>

typedef __attribute__((ext_vector_type(2))) float v2f;
typedef __attribute__((ext_vector_type(8))) float v8f;

#define C_DIM 256
#define HW 2304
#define W_DIM 48
#define N_INV 8
#define N_EQ 16
#define N_CP 16
#define N_ALL 40

// ---------------- Kernel 1: global average pool z[b,c] ----------------
__global__ void k_gap(const float* __restrict__ feat, float* __restrict__ z) {
    int bc = blockIdx.x;                       // b*256 + c
    const float* src = feat + (size_t)bc * HW;
    int t = threadIdx.x;
    float s = 0.f;
    #pragma unroll
    for (int i = 0; i < 9; ++i) s += src[t + i * 256];
    __shared__ float red[256];
    red[t] = s;
    __syncthreads();
    for (int off = 128; off > 0; off >>= 1) {
        if (t < off) red[t] += red[t + off];
        __syncthreads();
    }
    if (t == 0) z[bc] = red[0] * (1.0f / 2304.0f);
}

// ---------------- Kernel 2: channel sum fs[b,p] ----------------
__global__ void k_chansum(const float* __restrict__ feat, float* __restrict__ fs) {
    int b = blockIdx.y;
    int p = blockIdx.x * 256 + threadIdx.x;    // 9 x 256 = 2304
    const float* src = feat + (size_t)b * C_DIM * HW + p;
    float s = 0.f;
    for (int c = 0; c < C_DIM; ++c) s += src[(size_t)c * HW];
    fs[b * HW + p] = s;
}

// ---------------- Kernel 3: coefficient MLP (one branch per launch) ----------------
__global__ void k_coeff(const float* __restrict__ z, const float* __restrict__ w1,
                        const float* __restrict__ b1, const float* __restrict__ w2,
                        const float* __restrict__ b2, float* __restrict__ coeff, int nOut) {
    __shared__ float sz[256];
    __shared__ float sh[64];
    int b = blockIdx.x, t = threadIdx.x;
    sz[t] = z[b * 256 + t];
    __syncthreads();
    if (t < 64) {
        float acc = b1[t];
        for (int c = 0; c < 256; ++c) acc += sz[c] * w1[c * 64 + t];
        sh[t] = fmaxf(acc, 0.f);
    }
    __syncthreads();
    for (int j = t; j < nOut; j += 256) {
        float acc = b2[j];
        #pragma unroll 8
        for (int k = 0; k < 64; ++k) acc += sh[k] * w2[k * nOut + j];
        coeff[(size_t)b * nOut + j] = acc;      // layout [B, n, C] row-major ('bnc')
    }
}

// ---------------- Kernel 4: build all 40 basis kernels (7x7) on device ----------------
__global__ void k_basis(const float* __restrict__ rw1, const float* __restrict__ rb1,
                        const float* __restrict__ rw2, const float* __restrict__ rb2,
                        float* __restrict__ basis) {
    int n = threadIdx.x;
    if (n >= N_ALL) return;
    if (n < 24) {
        // Gaussian-derivative basis. combos[j] = (sigmas[j%3], orders[j/3]); i % 9 cycling.
        const bool is_inv = (n < 8);
        int i = is_inv ? n : (n - 8);
        int j = i % 9;
        float sig = is_inv ? ((j % 3 == 0) ? 4.f : (j % 3 == 1) ? 6.f : 8.f)
                           : ((j % 3 == 0) ? 1.f : (j % 3 == 1) ? 2.f : 3.f);
        int d = j / 3;
        float vals[49];
        float ss = 0.f;
        for (int t = 0; t < 49; ++t) {
            float cy = (float)(t / 7) - 3.f, cx = (float)(t % 7) - 3.f;
            float r = sqrtf(cx * cx + cy * cy + 1e-8f);
            float u = r / sig;
            float g = __expf(-0.5f * u * u);
            float p = (d == 0) ? 1.f : ((d == 1) ? u : (u * u - 1.f));
            float psi = p * g;
            vals[t] = psi;
            ss += psi * psi;
        }
        float inv_norm = 1.f / sqrtf(ss + 1e-8f);
        for (int t = 0; t < 49; ++t) basis[n * 49 + t] = vals[t] * inv_norm;
    } else {
        // Circular harmonic basis: basis[m*8 + a] = rad_m(r) * {cos,sin}((1+a/2)*theta)
        int m = (n - 24) / 8, a = (n - 24) % 8;
        float o = (float)(1 + a / 2);
        bool use_cos = ((a & 1) == 0);
        for (int t = 0; t < 49; ++t) {
            float cy = (float)(t / 7) - 3.f, cx = (float)(t % 7) - 3.f;
            float r = sqrtf(cx * cx + cy * cy + 1e-8f);
            float th = atan2f(cy, cx);
            float rad = rb2[m];
            for (int j2 = 0; j2 < 32; ++j2)
                rad += tanhf(r * rw1[j2] + rb1[j2]) * rw2[j2 * 2 + m];
            float ang = use_cos ? __cosf(o * th) : __sinf(o * th);
            basis[n * 49 + t] = rad * ang;
        }
    }
}

// ---------------- Kernel 5: basis convolution g[b,n,p] = fs[b] (*) basis[n] ----------------
__global__ void k_bconv(const float* __restrict__ fs, const float* __restrict__ basis,
                        float* __restrict__ gbuf) {
    __shared__ float sfs[HW];
    __shared__ float sk[49];
    int n = blockIdx.x, b = blockIdx.y, t = threadIdx.x;
    for (int i = t; i < HW; i += 256) sfs[i] = fs[b * HW + i];
    if (t < 49) sk[t] = basis[n * 49 + t];
    __syncthreads();
    #pragma unroll
    for (int i = 0; i < 9; ++i) {
        int p = t + i * 256;
        int y = p / W_DIM, x = p % W_DIM;
        float acc = 0.f;
        for (int dy = 0; dy < 7; ++dy) {
            int sy = y + dy - 3;
            if (sy < 0 || sy >= 48) continue;
            #pragma unroll
            for (int dx = 0; dx < 7; ++dx) {
                int sx = x + dx - 3;
                if (sx < 0 || sx >= 48) continue;
                acc += sfs[sy * W_DIM + sx] * sk[dy * 7 + dx];
            }
        }
        gbuf[((size_t)(b * N_ALL + n)) * HW + p] = acc;
    }
}

// ---------------- Branch GEMM + LayerNorm (compile-time K for full unroll) ----------------
// Wave w owns channel tiles {2w, 2w+1} (M=16 each); 3 pixel tiles (N=16) cover W=48.
// A fragments (coeff) hoisted to registers once per c-tile, reused across pixel tiles.
template <int NK>
__device__ __forceinline__ void branch_gemm_ln(
    const float* __restrict__ s_co,   // [NK][256] coeff slab for this branch
    const float* __restrict__ s_gb,   // [NK][48]  basis-conv row slab for this branch
    const float* __restrict__ lg, const float* __restrict__ lb,
    float* __restrict__ s_sum, float* __restrict__ s_sq,
    float* __restrict__ outp,         // out + br*outN
    int b, int y, int w, int half, int lm)
{
    constexpr int NQ = NK / 4;
    const v8f vzero = {0.f, 0.f, 0.f, 0.f, 0.f, 0.f, 0.f, 0.f};
    v8f acc[2][3];
    #pragma unroll
    for (int ci = 0; ci < 2; ++ci)
        #pragma unroll
        for (int pt = 0; pt < 3; ++pt) acc[ci][pt] = vzero;

    #pragma unroll
    for (int ci = 0; ci < 2; ++ci) {
        const int c0 = (2 * w + ci) * 16;
        v2f A[NQ];
        #pragma unroll
        for (int q = 0; q < NQ; ++q) {
            const int k0 = q * 4 + 2 * half;
            A[q].x = s_co[k0 * 256 + c0 + lm];
            A[q].y = s_co[(k0 + 1) * 256 + c0 + lm];
        }
        #pragma unroll
        for (int pt = 0; pt < 3; ++pt) {
            const int p0 = pt * 16;
            #pragma unroll
            for (int q = 0; q < NQ; ++q) {
                const int k0 = q * 4 + 2 * half;
                v2f B;
                B.x = s_gb[k0 * 48 + p0 + lm];
                B.y = s_gb[(k0 + 1) * 48 + p0 + lm];
                acc[ci][pt] = __builtin_amdgcn_wmma_f32_16x16x4_f32(
                    false, A[q], false, B, (short)0, acc[ci][pt], false, false);
            }
        }
    }

    // LayerNorm partial sums over channels (each lane: 8 channels x 1 pixel per tile)
    #pragma unroll
    for (int ci = 0; ci < 2; ++ci)
        #pragma unroll
        for (int pt = 0; pt < 3; ++pt) {
            float s = 0.f, q2 = 0.f;
            #pragma unroll
            for (int r = 0; r < 8; ++r) { float v = acc[ci][pt][r]; s += v; q2 += v * v; }
            atomicAdd(&s_sum[pt * 16 + lm], s);
            atomicAdd(&s_sq[pt * 16 + lm], q2);
        }
    __syncthreads();

    #pragma unroll
    for (int ci = 0; ci < 2; ++ci) {
        const int c0 = (2 * w + ci) * 16 + half * 8;
        #pragma unroll
        for (int pt = 0; pt < 3; ++pt) {
            const int p = pt * 16 + lm;
            const float mean = s_sum[p] * (1.f / 256.f);
            const float var = s_sq[p] * (1.f / 256.f) - mean * mean;
            const float rstd = rsqrtf(var + 1e-5f);
            #pragma unroll
            for (int r = 0; r < 8; ++r) {
                const int c = c0 + r;
                const float v = (acc[ci][pt][r] - mean) * rstd * lg[c] + lb[c];
                outp[(((size_t)(b * 256 + c)) * 48 + y) * 48 + p] = v;
            }
        }
    }
}

// ---------------- Kernel 6: fused WMMA GEMM (coeff x g) + LayerNorm + store ----------------
__global__ void k_main(const float* __restrict__ coeff_inv, const float* __restrict__ coeff_eq,
                       const float* __restrict__ coeff_cp, const float* __restrict__ gbuf,
                       const float* __restrict__ lg_inv, const float* __restrict__ lb_inv,
                       const float* __restrict__ lg_eq, const float* __restrict__ lb_eq,
                       const float* __restrict__ lg_cp, const float* __restrict__ lb_cp,
                       float* __restrict__ out) {
    __shared__ float s_coeff[N_ALL * 256];   // [n][c], inv@0, eq@2048, cp@6144
    __shared__ float s_g[N_ALL * 48];        // [n][x] for this row
    __shared__ float s_ln[6 * 256];          // inv_g,inv_b,eq_g,eq_b,cp_g,cp_b
    __shared__ float s_sum[48];
    __shared__ float s_sq[48];

    int y = blockIdx.x, b = blockIdx.y, t = threadIdx.x;

    for (int i = t; i < N_INV * 256; i += 256) s_coeff[i]        = coeff_inv[(size_t)b * N_INV * 256 + i];
    for (int i = t; i < N_EQ * 256;  i += 256) s_coeff[2048 + i] = coeff_eq[(size_t)b * N_EQ * 256 + i];
    for (int i = t; i < N_CP * 256;  i += 256) s_coeff[6144 + i] = coeff_cp[(size_t)b * N_CP * 256 + i];
    for (int i = t; i < N_ALL * 48; i += 256) {
        int n = i / 48, x = i % 48;
        s_g[i] = gbuf[((size_t)(b * N_ALL + n)) * HW + y * W_DIM + x];
    }
    s_ln[t]        = lg_inv[t];  s_ln[256 + t]  = lb_inv[t];
    s_ln[512 + t]  = lg_eq[t];   s_ln[768 + t]  = lb_eq[t];
    s_ln[1024 + t] = lg_cp[t];   s_ln[1280 + t] = lb_cp[t];
    __syncthreads();

    const int w = t >> 5, lane = t & 31;
    const int half = lane >> 4, lm = lane & 15;
    const size_t outN = (size_t)16 * 256 * 48 * 48;

    // branch 0: inv (K = 8)
    if (t < 48) { s_sum[t] = 0.f; s_sq[t] = 0.f; }
    __syncthreads();
    branch_gemm_ln<N_INV>(s_coeff, s_g, &s_ln[0], &s_ln[256],
                          s_sum, s_sq, out, b, y, w, half, lm);
    __syncthreads();

    // branch 1: eq (K = 16)
    if (t < 48) { s_sum[t] = 0.f; s_sq[t] = 0.f; }
    __syncthreads();
    branch_gemm_ln<N_EQ>(s_coeff + 2048, s_g + N_INV * 48, &s_ln[512], &s_ln[768],
                         s_sum, s_sq, out + outN, b, y, w, half, lm);
    __syncthreads();

    // branch 2: cp (K = 16)
    if (t < 48) { s_sum[t] = 0.f; s_sq[t] = 0.f; }
    __syncthreads();
    branch_gemm_ln<N_CP>(s_coeff + 6144, s_g + (N_INV + N_EQ) * 48, &s_ln[1024], &s_ln[1280],
                         s_sum, s_sq, out + 2 * outN, b, y, w, half, lm);
}

extern "C" void kernel_launch(void* const* d_in, const int* in_sizes, int n_in,
                              void* d_out, int out_size, void* d_ws, size_t ws_size,
                              hipStream_t stream) {
    (void)in_sizes; (void)n_in; (void)out_size; (void)ws_size;
    const float* feature = (const float*)d_in[0];
    const float* iw1 = (const float*)d_in[1];  const float* ib1 = (const float*)d_in[2];
    const float* iw2 = (const float*)d_in[3];  const float* ib2 = (const float*)d_in[4];
    const float* ew1 = (const float*)d_in[5];  const float* eb1 = (const float*)d_in[6];
    const float* ew2 = (const float*)d_in[7];  const float* eb2 = (const float*)d_in[8];
    const float* pw1 = (const float*)d_in[9];  const float* pb1 = (const float*)d_in[10];
    const float* pw2 = (const float*)d_in[11]; const float* pb2 = (const float*)d_in[12];
    const float* rw1 = (const float*)d_in[13]; const float* rb1 = (const float*)d_in[14];
    const float* rw2 = (const float*)d_in[15]; const float* rb2 = (const float*)d_in[16];
    const float* lg_inv = (const float*)d_in[17]; const float* lb_inv = (const float*)d_in[18];
    const float* lg_eq  = (const float*)d_in[19]; const float* lb_eq  = (const float*)d_in[20];
    const float* lg_cp  = (const float*)d_in[21]; const float* lb_cp  = (const float*)d_in[22];
    float* out = (float*)d_out;

    // Workspace layout (floats): total ~1.69M floats (~6.8 MB)
    float* ws    = (float*)d_ws;
    float* z     = ws;                 // 4096
    float* fs    = ws + 4096;          // 36864
    float* basis = ws + 40960;         // 1960
    float* c_inv = ws + 43008;         // 32768
    float* c_eq  = ws + 75776;         // 65536
    float* c_cp  = ws + 141312;        // 65536
    float* gbuf  = ws + 206848;        // 1474560

    k_gap<<<dim3(16 * 256), 256, 0, stream>>>(feature, z);
    k_chansum<<<dim3(9, 16), 256, 0, stream>>>(feature, fs);
    k_coeff<<<dim3(16), 256, 0, stream>>>(z, iw1, ib1, iw2, ib2, c_inv, N_INV * 256);
    k_coeff<<<dim3(16), 256, 0, stream>>>(z, ew1, eb1, ew2, eb2, c_eq, N_EQ * 256);
    k_coeff<<<dim3(16), 256, 0, stream>>>(z, pw1, pb1, pw2, pb2, c_cp, N_CP * 256);
    k_basis<<<dim3(1), 64, 0, stream>>>(rw1, rb1, rw2, rb2, basis);
    k_bconv<<<dim3(N_ALL, 16), 256, 0, stream>>>(fs, basis, gbuf);
    k_main<<<dim3(48, 16), 256, 0, stream>>>(c_inv, c_eq, c_cp, gbuf,
                                             lg_inv, lb_inv, lg_eq, lb_eq, lg_cp, lb_cp, out);
}